// TCSAModule_63745904608082
// MI455X (gfx1250) — compile-verified
//
#include <hip/hip_runtime.h>
#include <math.h>

#define T_  12
#define N_  32
#define C_  128
#define HW_ 1024
#define R_  64

#define SCS_STRIDE 68   // padded sc_s row stride (floats): conflict-free b64 A loads
#define BS_STRIDE  160  // padded bs2 q stride (floats): conflict-free b64 B loads

typedef __attribute__((ext_vector_type(2))) float v2f;
typedef __attribute__((ext_vector_type(8))) float v8f;

// ---------------------------------------------------------------------------
// Kernel 1: one pass over x[t][n][:][:] producing
//   ts_in[n][t][p]  = mean over c   (per-thread accumulator, p = tid)
//   pooled[n][t][c] = mean over p   (wave32 shfl tree + LDS cross-wave)
// ---------------------------------------------------------------------------
__global__ void k_reduce(const float* __restrict__ x,
                         float* __restrict__ ts_in,
                         float* __restrict__ pooled) {
    const int t = blockIdx.x / N_;
    const int n = blockIdx.x % N_;
    const int tid  = threadIdx.x;      // 0..1023, p = tid
    const int wave = tid >> 5;
    const int lane = tid & 31;

    __shared__ float part[32 * C_];    // 16 KB: per-wave partial sums per c

    const float* xb = x + ((size_t)(t * N_ + n)) * C_ * HW_;
    float tsacc = 0.0f;

    for (int c = 0; c < C_; ++c) {
        float v = xb[(size_t)c * HW_ + tid];
        tsacc += v;
        float s = v;                   // wave32 butterfly reduction
        s += __shfl_xor(s, 16);
        s += __shfl_xor(s, 8);
        s += __shfl_xor(s, 4);
        s += __shfl_xor(s, 2);
        s += __shfl_xor(s, 1);
        if (lane == 0) part[wave * C_ + c] = s;
    }

    ts_in[(n * T_ + t) * HW_ + tid] = tsacc * (1.0f / C_);

    __syncthreads();
    if (tid < C_) {
        float s = 0.0f;
        #pragma unroll
        for (int w = 0; w < 32; ++w) s += part[w * C_ + tid];
        pooled[(n * T_ + t) * C_ + tid] = s * (1.0f / HW_);
    }
}

// ---------------------------------------------------------------------------
// Kernel 2: softmax over the 64 rank weights (single block)
// ---------------------------------------------------------------------------
__global__ void k_softmax(const float* __restrict__ w_rank,
                          float* __restrict__ wr) {
    const int tid = threadIdx.x;  // 64 threads
    float m = -1e30f;
    for (int i = 0; i < R_; ++i) m = fmaxf(m, w_rank[i]);
    float s = 0.0f;
    for (int i = 0; i < R_; ++i) s += expf(w_rank[i] - m);
    wr[tid] = expf(w_rank[tid] - m) / s;
}

// ---------------------------------------------------------------------------
// Kernel 3: tc[n][t][r] = sigmoid(pooled[n][t][:] . tc_w[r][:] + tc_b[r])
// ---------------------------------------------------------------------------
__global__ void k_tc(const float* __restrict__ pooled,
                     const float* __restrict__ tc_w,
                     const float* __restrict__ tc_b,
                     float* __restrict__ tco) {
    const int o = blockIdx.x * blockDim.x + threadIdx.x;  // N*T*R
    const int n = o / (T_ * R_);
    const int t = (o / R_) % T_;
    const int r = o % R_;
    const float* pr = pooled + (n * T_ + t) * C_;
    const float* wrow = tc_w + r * C_;
    float s = tc_b[r];
    for (int c = 0; c < C_; ++c) s += pr[c] * wrow[c];
    tco[(n * T_ + t) * R_ + r] = 1.0f / (1.0f + expf(-s));
}

// ---------------------------------------------------------------------------
// Kernel 4: sc[n][c][r] = sigmoid(sum_t pooled[n][t][c]*sc_w[r][t] + sc_b[r])
// ---------------------------------------------------------------------------
__global__ void k_sc(const float* __restrict__ pooled,
                     const float* __restrict__ sc_w,
                     const float* __restrict__ sc_b,
                     float* __restrict__ sco) {
    const int o = blockIdx.x * blockDim.x + threadIdx.x;  // N*C*R
    const int n = o / (C_ * R_);
    const int c = (o / R_) % C_;
    const int r = o % R_;
    float s = sc_b[r];
    #pragma unroll
    for (int t = 0; t < T_; ++t)
        s += pooled[(n * T_ + t) * C_ + c] * sc_w[r * T_ + t];
    sco[(n * C_ + c) * R_ + r] = 1.0f / (1.0f + expf(-s));
}

// ---------------------------------------------------------------------------
// Kernel 5: 3x3 SAME conv T->R over ts_in, + bias, sigmoid.
// Block = (r, n); ts_in slab (48 KB, b128 fills) + filter row staged in LDS.
// ---------------------------------------------------------------------------
__global__ void k_conv(const float* __restrict__ ts_in,
                       const float* __restrict__ conv_w,
                       const float* __restrict__ conv_b,
                       float* __restrict__ ts) {
    const int r = blockIdx.x;
    const int n = blockIdx.y;
    const int tid = threadIdx.x;  // 256

    __shared__ float tss[T_ * HW_];   // 48 KB
    __shared__ float cw[T_ * 9];

    {
        const float4* src = (const float4*)(ts_in + (size_t)n * T_ * HW_);
        float4* dst = (float4*)tss;
        for (int i = tid; i < (T_ * HW_) / 4; i += 256) dst[i] = src[i];
    }
    for (int i = tid; i < T_ * 9; i += 256) cw[i] = conv_w[r * T_ * 9 + i];
    __syncthreads();

    const float b = conv_b[r];
    #pragma unroll
    for (int j = 0; j < 4; ++j) {
        const int p = tid + j * 256;
        const int h = p >> 5, w = p & 31;
        float s = b;
        for (int t = 0; t < T_; ++t) {
            #pragma unroll
            for (int dy = 0; dy < 3; ++dy) {
                const int hy = h + dy - 1;
                if ((unsigned)hy >= 32u) continue;
                #pragma unroll
                for (int dx = 0; dx < 3; ++dx) {
                    const int wx = w + dx - 1;
                    if ((unsigned)wx >= 32u) continue;
                    s += tss[t * HW_ + hy * 32 + wx] * cw[t * 9 + dy * 3 + dx];
                }
            }
        }
        ts[(size_t)(n * R_ + r) * HW_ + p] = 1.0f / (1.0f + expf(-s));
    }
}

// ---------------------------------------------------------------------------
// Kernel 6: fused rank-64 CP reconstruction + relu(result*x), fp32 WMMA.
// Per n: D[(t,c), p] = sum_r (wr[r]*tc[n,t,r]*sc[n,c,r]) * ts[n,r,p]
// Block = (p-tile of 64, t, n); 8 waves, wave w -> c-block w, 4 p-subtiles.
//
// LDS layouts chosen for conflict-free b64 fragment loads:
//  - sc_s rows padded to 68 floats: half-waves land on disjoint bank-pair sets.
//  - bs2 is K-pair interleaved: bs2[q*160 + j*2 + par] = ts[2q+par][p0+j],
//    so one aligned b64 load IS the v2f B fragment (no register repacking),
//    and the two half-waves (q parity) use disjoint bank pairs.
// A fragment (16x4 fp32): lane<16 holds K=k0,k0+1; lane>=16 holds K=k0+2,k0+3.
// D: VGPR i -> (M = i + (lane>=16?8:0), N = col).
// ---------------------------------------------------------------------------
__global__ void k_fuse(const float* __restrict__ x,
                       const float* __restrict__ wr,
                       const float* __restrict__ tc,
                       const float* __restrict__ sc,
                       const float* __restrict__ ts,
                       float* __restrict__ out) {
    const int p0 = blockIdx.x * 64;
    const int t  = blockIdx.y;
    const int n  = blockIdx.z;
    const int tid  = threadIdx.x;   // 256
    const int wv   = tid >> 5;      // 0..7  -> c block
    const int lane = tid & 31;

    __shared__ float a_tc[R_];                    // wr[r]*tc[n][t][r]
    __shared__ float sc_s[C_ * SCS_STRIDE];       // ~34 KB, padded rows
    __shared__ float bs2[32 * BS_STRIDE];         // ~20 KB, K-pair interleaved

    if (tid < R_) a_tc[tid] = wr[tid] * tc[(n * T_ + t) * R_ + tid];
    {   // sc[n][c][0..63] -> sc_s[c*68 + 0..63], b128 both sides
        const float4* src = (const float4*)(sc + (size_t)n * C_ * R_);
        for (int i = tid; i < (C_ * R_) / 4; i += 256) {
            const int cc = i >> 4, v = i & 15;
            *(float4*)(sc_s + cc * SCS_STRIDE + v * 4) = src[i];
        }
    }
    {   // ts[n][2q..2q+1][p0+4j4..+3] -> interleaved pairs, b128 both sides
        for (int i = tid; i < 512; i += 256) {
            const int q = i >> 4, j4 = i & 15;
            const float* r0 = ts + (size_t)(n * R_ + 2 * q) * HW_ + p0 + j4 * 4;
            const float4 f0 = *(const float4*)r0;
            const float4 f1 = *(const float4*)(r0 + HW_);
            float* d = bs2 + q * BS_STRIDE + j4 * 8;
            *(float4*)(d)     = make_float4(f0.x, f1.x, f0.y, f1.y);
            *(float4*)(d + 4) = make_float4(f0.z, f1.z, f0.w, f1.w);
        }
    }
    __syncthreads();

    const int row = lane & 15;            // A row / B,D column
    const int hi  = lane >> 4;            // half-wave select
    const int kb  = hi * 2;               // per-half-wave K sub-offset
    const int c   = wv * 16 + row;

    const float* atl = a_tc + kb;
    const float* scl = sc_s + c * SCS_STRIDE + kb;
    const float* bsl = bs2 + hi * BS_STRIDE + row * 2;

    v8f acc[4] = {v8f{}, v8f{}, v8f{}, v8f{}};
    #pragma unroll
    for (int kk = 0; kk < 16; ++kk) {
        const v2f av = *(const v2f*)(atl + kk * 4);
        const v2f sv = *(const v2f*)(scl + kk * 4);
        const v2f a  = av * sv;                       // v_pk_mul_f32
        #pragma unroll
        for (int pb = 0; pb < 4; ++pb) {
            const v2f b = *(const v2f*)(bsl + kk * 2 * BS_STRIDE + pb * 32);
            acc[pb] = __builtin_amdgcn_wmma_f32_16x16x4_f32(
                /*neg_a=*/false, a, /*neg_b=*/false, b,
                /*c_mod=*/(short)0, acc[pb], /*reuse_a=*/false, /*reuse_b=*/false);
        }
    }

    // Fused epilogue: out = relu(D * x), two 64B row segments per store op.
    const int mofs = hi * 8;
    const size_t base = ((size_t)(t * N_ + n) * C_) * HW_;
    #pragma unroll
    for (int pb = 0; pb < 4; ++pb) {
        #pragma unroll
        for (int i = 0; i < 8; ++i) {
            const int cc = wv * 16 + i + mofs;
            const size_t idx = base + (size_t)cc * HW_ + p0 + pb * 16 + row;
            const float xv = x[idx];
            const float v  = acc[pb][i] * xv;
            out[idx] = v > 0.0f ? v : 0.0f;
        }
    }
}

// ---------------------------------------------------------------------------
// Host launcher
// ---------------------------------------------------------------------------
extern "C" void kernel_launch(void* const* d_in, const int* in_sizes, int n_in,
                              void* d_out, int out_size, void* d_ws, size_t ws_size,
                              hipStream_t stream) {
    (void)in_sizes; (void)n_in; (void)out_size; (void)ws_size;
    const float* x      = (const float*)d_in[0];
    const float* w_rank = (const float*)d_in[1];
    const float* conv_w = (const float*)d_in[2];
    const float* conv_b = (const float*)d_in[3];
    const float* tc_w   = (const float*)d_in[4];
    const float* tc_b   = (const float*)d_in[5];
    const float* sc_w   = (const float*)d_in[6];
    const float* sc_b   = (const float*)d_in[7];
    float* out = (float*)d_out;
    float* ws  = (float*)d_ws;

    // workspace layout (floats): total ~11.3 MB
    float* ts_in  = ws;               // N*T*HW   = 393216
    float* pooled = ws + 393216;      // N*T*C    =  49152
    float* wr     = ws + 442368;      // R        =     64
    float* tcb    = ws + 442432;      // N*T*R    =  24576
    float* scb    = ws + 467008;      // N*C*R    = 262144
    float* tsb    = ws + 729152;      // N*R*HW   = 2097152

    k_reduce <<<dim3(T_ * N_), dim3(1024), 0, stream>>>(x, ts_in, pooled);
    k_softmax<<<1, R_, 0, stream>>>(w_rank, wr);
    k_tc     <<<dim3((N_ * T_ * R_) / 256), 256, 0, stream>>>(pooled, tc_w, tc_b, tcb);
    k_sc     <<<dim3((N_ * C_ * R_) / 256), 256, 0, stream>>>(pooled, sc_w, sc_b, scb);
    k_conv   <<<dim3(R_, N_), 256, 0, stream>>>(ts_in, conv_w, conv_b, tsb);
    k_fuse   <<<dim3(HW_ / 64, T_, N_), 256, 0, stream>>>(x, wr, tcb, scb, tsb, out);
}